// ChamferDistanceLoss_84945863180902
// MI455X (gfx1250) — compile-verified
//
#include <hip/hip_runtime.h>
#include <cstdint>

typedef __attribute__((ext_vector_type(2))) float v2f;
typedef __attribute__((ext_vector_type(8))) float v8f;

#define N_PTS 65536
#define M_PTS 8192
#define ROWS_PER_BLOCK 128
#define THREADS 256
#define TILES_M (M_PTS / 16)
#define INF_BITS 0x7F800000u

// ---------------------------------------------------------------------------
// Prep: y4[m] = (-2*y0, -2*y1, -2*y2, ||y||^2); init global column-min to +inf
// ---------------------------------------------------------------------------
__global__ void chamfer_prep(const float* __restrict__ y,
                             float4* __restrict__ y4,
                             unsigned* __restrict__ colMinBits) {
    int m = blockIdx.x * blockDim.x + threadIdx.x;
    if (m < M_PTS) {
        float y0 = y[3 * m + 0];
        float y1 = y[3 * m + 1];
        float y2 = y[3 * m + 2];
        float4 q;
        q.x = -2.0f * y0;
        q.y = -2.0f * y1;
        q.z = -2.0f * y2;
        q.w = y0 * y0 + y1 * y1 + y2 * y2;
        y4[m] = q;
        colMinBits[m] = INF_BITS;
    }
}

// ---------------------------------------------------------------------------
// Main: each wave owns 16 rows (N side), sweeps all M in 16-col WMMA tiles.
//   d2[n,m] = max( (x|x2) . (-2y|1) + y2 , 0 )  via V_WMMA_F32_16X16X4_F32
// ---------------------------------------------------------------------------
__launch_bounds__(THREADS)
__global__ void chamfer_main(const float* __restrict__ x,
                             const float4* __restrict__ y4,
                             const float* __restrict__ probs,
                             unsigned* __restrict__ colMinBits,
                             float* __restrict__ l2partial) {
    __shared__ unsigned colMinLds[M_PTS];   // 32 KB block-local column mins
    __shared__ float red[THREADS];

    const int tid  = threadIdx.x;
    const int lane = tid & 31;
    const int wave = tid >> 5;
    const int slot = lane & 15;             // column slot within a tile
    const bool hi  = (lane >= 16);

    for (int i = tid; i < M_PTS; i += THREADS) colMinLds[i] = INF_BITS;
    __syncthreads();

    // ---- A fragment: 16x4 f32. lanes 0-15: K=0,1 ; lanes 16-31: K=2,3 ----
    const int row = blockIdx.x * ROWS_PER_BLOCK + wave * 16 + slot;
    const float x0 = x[3 * row + 0];
    const float x1 = x[3 * row + 1];
    const float x2 = x[3 * row + 2];
    const float xs = x0 * x0 + x1 * x1 + x2 * x2;
    v2f a;
    a.x = hi ? x2 : x0;   // K=2 : K=0
    a.y = hi ? xs : x1;   // K=3 : K=1

    float best[8], bestp[8];
#pragma unroll
    for (int v = 0; v < 8; ++v) { best[v] = __builtin_inff(); bestp[v] = 0.0f; }

    for (int t = 0; t < TILES_M; ++t) {
        const int idx = t * 16 + slot;
        const float4 q = y4[idx];
        const float  p = probs[idx];

        // ---- B fragment: 4x16 f32, mirrored layout ----
        v2f b;
        b.x = hi ? q.z  : q.x;    // K=2 : K=0
        b.y = hi ? 1.0f : q.y;    // K=3 : K=1

        v8f c = {};
        c = __builtin_amdgcn_wmma_f32_16x16x4_f32(
                /*neg_a=*/false, a, /*neg_b=*/false, b,
                /*c_mod=*/(short)0, c, /*reuse_a=*/false, /*reuse_b=*/false);

        float cmin = __builtin_inff();
#pragma unroll
        for (int v = 0; v < 8; ++v) {
            float d = fmaxf(c[v] + q.w, 0.0f);    // add ||y||^2, clamp
            cmin = fminf(cmin, d);                // column-min over 8 rows
            bool lt = d < best[v];                // row running min + its prob
            bestp[v] = lt ? p : bestp[v];
            best[v]  = lt ? d : best[v];
        }

        // fold the two half-waves (same column), then block-local LDS min
        float other = __shfl_xor(cmin, 16, 32);
        cmin = fminf(cmin, other);
        if (!hi) atomicMin(&colMinLds[idx], __float_as_uint(cmin));
    }

    // ---- flush block-local column mins to global (uint min == f32 min >=0) --
    __syncthreads();
    for (int i = tid; i < M_PTS; i += THREADS)
        atomicMin(&colMinBits[i], colMinLds[i]);

    // ---- row min/argmin: butterfly within each 16-lane half ----
#pragma unroll
    for (int v = 0; v < 8; ++v) {
#pragma unroll
        for (int m = 1; m < 16; m <<= 1) {
            float ob = __shfl_xor(best[v], m, 32);
            float op = __shfl_xor(bestp[v], m, 32);
            bool lt = ob < best[v];
            best[v]  = lt ? ob : best[v];
            bestp[v] = lt ? op : bestp[v];
        }
    }
    float local = 0.0f;
    if (lane == 0 || lane == 16) {        // lane0: rows v ; lane16: rows v+8
#pragma unroll
        for (int v = 0; v < 8; ++v) local += best[v] * bestp[v];
    }

    // ---- deterministic block tree-reduce of l2 contributions ----
    red[tid] = local;
    __syncthreads();
    for (int s = THREADS / 2; s > 0; s >>= 1) {
        if (tid < s) red[tid] += red[tid + s];
        __syncthreads();
    }
    if (tid == 0) l2partial[blockIdx.x] = red[0];
}

// ---------------------------------------------------------------------------
// Final: l1 = sum_m probs[m]*colmin[m] ; l2 = sum_b partial[b] ; out = l1+l2
// Fixed-order sums -> bit-deterministic across replays.
// ---------------------------------------------------------------------------
__global__ void chamfer_final(const unsigned* __restrict__ colMinBits,
                              const float* __restrict__ probs,
                              const float* __restrict__ l2partial,
                              int nblocks,
                              float* __restrict__ out) {
    __shared__ float red[THREADS];
    const int tid = threadIdx.x;
    float s = 0.0f;
    for (int m = tid; m < M_PTS; m += THREADS)
        s += probs[m] * __uint_as_float(colMinBits[m]);
    for (int b = tid; b < nblocks; b += THREADS)
        s += l2partial[b];
    red[tid] = s;
    __syncthreads();
    for (int k = THREADS / 2; k > 0; k >>= 1) {
        if (tid < k) red[tid] += red[tid + k];
        __syncthreads();
    }
    if (tid == 0) out[0] = red[0];
}

// ---------------------------------------------------------------------------
extern "C" void kernel_launch(void* const* d_in, const int* in_sizes, int n_in,
                              void* d_out, int out_size, void* d_ws, size_t ws_size,
                              hipStream_t stream) {
    (void)in_sizes; (void)n_in; (void)out_size; (void)ws_size;

    const float* x     = (const float*)d_in[0];   // [N,3] input_vertex_set
    const float* y     = (const float*)d_in[1];   // [M,3] sampled_points
    const float* probs = (const float*)d_in[2];   // [M]
    float* out = (float*)d_out;

    char* ws = (char*)d_ws;
    unsigned* colMinBits = (unsigned*)ws;                               // 32 KB
    float4*   y4         = (float4*)(ws + M_PTS * sizeof(unsigned));   // 128 KB
    float*    l2partial  = (float*)(ws + M_PTS * sizeof(unsigned)
                                       + M_PTS * sizeof(float4));      //  2 KB

    const int nblocks = N_PTS / ROWS_PER_BLOCK;   // 512

    chamfer_prep <<<M_PTS / THREADS, THREADS, 0, stream>>>(y, y4, colMinBits);
    chamfer_main <<<nblocks,        THREADS, 0, stream>>>(x, y4, probs,
                                                          colMinBits, l2partial);
    chamfer_final<<<1,              THREADS, 0, stream>>>(colMinBits, probs,
                                                          l2partial, nblocks, out);
}